// Model_343597383985
// MI455X (gfx1250) — compile-verified
//
#include <hip/hip_runtime.h>

// ---------------------------------------------------------------------------
// Soft silhouette render + per-row MSE loss for MI455X (gfx1250, wave32).
//
// Edge-distance evaluation mapped onto V_WMMA_F32_16X16X4_F32:
//   D_e[16px x 16faces] = A[16x4 = (px,py,1,0)] x B_e[4x16 = (cx,cy,c0,0)]
// Three WMMAs (one per triangle edge) share the C/D register layout, so the
// 3-edge min is elementwise v_min3 on the v8f accumulators. Softplus uses raw
// base-2 HW transcendentals (v_exp_f32/v_log_f32 via amdgcn builtins, no
// denormal guards) with the ln2 scale hoisted out of the face loop via a
// split accumulator. 16-lane shuffle reduction gives per-pixel face sums.
// ---------------------------------------------------------------------------

typedef __attribute__((ext_vector_type(2))) float v2f;
typedef __attribute__((ext_vector_type(8))) float v8f;

#define IMG   256
#define NV    1000
#define NF    1000
#define NFP   1024          // faces padded to 64 tiles of 16
#define EPSF  1e-8f
#define LOG2E 1.4426950408889634f
#define LN2   0.6931471805599453f

// workspace layout (floats):
//   [0      , 2048 )  verts2d   (1024 x float2)
//   [2048   , 8192 )  cxy[e=0..2] each 1024 x float2  (cx, cy)
//   [8192   , 14336)  c0z[e=0..2] each 1024 x float2  (c0, 0)
//   [14336  , 18432)  tileErr   (4096 floats, one per 16-pixel tile)
#define WS_V2D   0
#define WS_CXY   2048
#define WS_C0Z   8192
#define WS_TERR  14336

// --------------------------- vertex transform ------------------------------
__global__ void k_transform(const float* __restrict__ verts,
                            const float* __restrict__ q,
                            const float* __restrict__ t,
                            const float* __restrict__ K,
                            float* __restrict__ v2d) {
    int v = blockIdx.x * blockDim.x + threadIdx.x;
    if (v >= NV) return;
    float qw = q[0], qx = q[1], qy = q[2], qz = q[3];
    float inv = 1.0f / sqrtf(qw*qw + qx*qx + qy*qy + qz*qz + EPSF);
    qw *= inv; qx *= inv; qy *= inv; qz *= inv;
    float r00 = 1.f - 2.f*(qy*qy + qz*qz), r01 = 2.f*(qx*qy - qw*qz), r02 = 2.f*(qx*qz + qw*qy);
    float r10 = 2.f*(qx*qy + qw*qz), r11 = 1.f - 2.f*(qx*qx + qz*qz), r12 = 2.f*(qy*qz - qw*qx);
    float r20 = 2.f*(qx*qz - qw*qy), r21 = 2.f*(qy*qz + qw*qx), r22 = 1.f - 2.f*(qx*qx + qy*qy);
    float vx = verts[3*v], vy = verts[3*v+1], vz = verts[3*v+2];
    float cx = r00*vx + r01*vy + r02*vz + t[0];
    float cy = r10*vx + r11*vy + r12*vz + t[1];
    float cz = r20*vx + r21*vy + r22*vz + t[2];
    float z  = cz + EPSF;
    v2d[2*v]   = K[0]*cx/z + K[2];
    v2d[2*v+1] = K[4]*cy/z + K[5];
}

// --------------------------- edge coefficients -----------------------------
__global__ void k_edges(const int*   __restrict__ faces,
                        const float* __restrict__ v2d,
                        float*       __restrict__ ws) {
    int f = blockIdx.x * blockDim.x + threadIdx.x;
    if (f >= NFP) return;

    float2* cxyA[3] = { (float2*)(ws + WS_CXY),
                        (float2*)(ws + WS_CXY + 2048),
                        (float2*)(ws + WS_CXY + 4096) };
    float2* c0zA[3] = { (float2*)(ws + WS_C0Z),
                        (float2*)(ws + WS_C0Z + 2048),
                        (float2*)(ws + WS_C0Z + 4096) };

    if (f >= NF) {  // padding: d = -1e4 -> softplus contribution = 0
        for (int e = 0; e < 3; ++e) {
            cxyA[e][f] = make_float2(0.f, 0.f);
            c0zA[e][f] = make_float2(-1.0e4f, 0.f);
        }
        return;
    }
    int i0 = faces[3*f], i1 = faces[3*f+1], i2 = faces[3*f+2];
    float tx[3] = { v2d[2*i0],   v2d[2*i1],   v2d[2*i2]   };
    float ty[3] = { v2d[2*i0+1], v2d[2*i1+1], v2d[2*i2+1] };
    float e01x = tx[1]-tx[0], e01y = ty[1]-ty[0];
    float e02x = tx[2]-tx[0], e02y = ty[2]-ty[0];
    float area2 = e01x*e02y - e01y*e02x;
    float s = (area2 >= 0.f) ? 1.0f : -1.0f;
    #pragma unroll
    for (int k = 0; k < 3; ++k) {
        int kn = (k + 1) % 3;
        float ax = tx[k], ay = ty[k];
        float ex = tx[kn] - ax, ey = ty[kn] - ay;
        float inv_len = s / sqrtf(ex*ex + ey*ey + EPSF);
        float cx = -ey * inv_len;
        float cy =  ex * inv_len;
        float c0 = (ey*ax - ex*ay) * inv_len;
        cxyA[k][f] = make_float2(cx, cy);
        c0zA[k][f] = make_float2(c0, 0.f);
    }
}

// ------------------------------- render ------------------------------------
// grid = 4096 (one 16-pixel tile of one image row), block = 128 (4 waves).
__global__ void __launch_bounds__(128)
k_render(const float* __restrict__ ws,
         const float* __restrict__ image_ref,
         float*       __restrict__ tileErr) {
    const int tile    = blockIdx.x;         // 0..4095
    const int row     = tile >> 4;          // image row i
    const int colbase = (tile & 15) << 4;   // first column j of this tile
    const int tid     = threadIdx.x;
    const int wave    = tid >> 5;
    const int lane    = tid & 31;
    const int li      = lane & 15;
    const bool hi     = lane >= 16;

    const float2* cxy[3];
    const float2* c0z[3];
    #pragma unroll
    for (int e = 0; e < 3; ++e) {
        cxy[e] = (const float2*)(ws + WS_CXY + e*2048);
        c0z[e] = (const float2*)(ws + WS_C0Z + e*2048);
    }

    // A operand, 16x4 f32: lanes 0-15 -> {px,py} (K=0,1) for pixel M=li;
    //                      lanes 16-31 -> {1,0}   (K=2,3)
    v2f a;
    if (!hi) { a.x = (float)(colbase + li) + 0.5f; a.y = (float)row + 0.5f; }
    else     { a.x = 1.0f;                         a.y = 0.0f;              }

    v8f accLin = {};   // sum of max(d, 0)          (ln-domain, exact)
    v8f accL2  = {};   // sum of log2(1 + 2^(-|d|*log2e)); scaled by ln2 later
    v8f zeroc  = {};

    // 64 face tiles of 16, 16 tiles per wave
    for (int tf = wave; tf < NFP/16; tf += 4) {
        const int fb = tf << 4;
        // B operand, 4x16 f32: lanes 0-15 -> {cx,cy} for face N=li;
        //                      lanes 16-31 -> {c0,0}
        float2 b0r = (hi ? c0z[0] : cxy[0])[fb + li];
        float2 b1r = (hi ? c0z[1] : cxy[1])[fb + li];
        float2 b2r = (hi ? c0z[2] : cxy[2])[fb + li];
        v2f b0; b0.x = b0r.x; b0.y = b0r.y;
        v2f b1; b1.x = b1r.x; b1.y = b1r.y;
        v2f b2; b2.x = b2r.x; b2.y = b2r.y;

        v8f d0 = __builtin_amdgcn_wmma_f32_16x16x4_f32(false, a, false, b0,
                                                       (short)0, zeroc, false, false);
        v8f d1 = __builtin_amdgcn_wmma_f32_16x16x4_f32(false, a, false, b1,
                                                       (short)0, zeroc, false, false);
        v8f d2 = __builtin_amdgcn_wmma_f32_16x16x4_f32(false, a, false, b2,
                                                       (short)0, zeroc, false, false);
        #pragma unroll
        for (int r = 0; r < 8; ++r) {
            float d = fminf(fminf(d0[r], d1[r]), d2[r]);
            // stable softplus, base-2 HW transcendentals, no denorm guards:
            // log1p(exp(d)) = max(d,0) + ln2 * log2(1 + 2^(-|d|*log2e))
            // argument of log2 is in (1,2] -> raw v_log_f32 is safe & accurate
            float t2 = __builtin_amdgcn_exp2f(-fabsf(d) * LOG2E);
            accL2[r]  += __builtin_amdgcn_logf(1.0f + t2);
            accLin[r] += fmaxf(d, 0.f);
        }
    }

    // combine ln-domain pieces, then reduce across each 16-lane half
    #pragma unroll
    for (int r = 0; r < 8; ++r) {
        float x = accLin[r] + LN2 * accL2[r];
        x += __shfl_xor(x, 1, 32);
        x += __shfl_xor(x, 2, 32);
        x += __shfl_xor(x, 4, 32);
        x += __shfl_xor(x, 8, 32);
        accLin[r] = x;
    }

    __shared__ float partial[4][16];
    __shared__ float errsh[16];
    if (li == 0) {
        // lane 0 holds pixels M=0..7, lane 16 holds pixels M=8..15
        #pragma unroll
        for (int r = 0; r < 8; ++r)
            partial[wave][(hi ? 8 : 0) + r] = accLin[r];
    }
    __syncthreads();

    if (tid < 16) {
        float S = partial[0][tid] + partial[1][tid] + partial[2][tid] + partial[3][tid];
        float sil  = 1.0f - __builtin_amdgcn_exp2f(-S * LOG2E);  // 1 - exp(-S)
        float ref  = image_ref[row*IMG + colbase + tid];
        float diff = sil - ref;
        errsh[tid] = diff * diff;
    }
    __syncthreads();

    if (tid == 0) {
        float ssum = 0.f;
        #pragma unroll
        for (int k = 0; k < 16; ++k) ssum += errsh[k];
        tileErr[tile] = ssum;
    }
}

// ------------------------------ finalize -----------------------------------
__global__ void k_finalize(const float* __restrict__ tileErr,
                           float*       __restrict__ out) {
    int i = blockIdx.x * blockDim.x + threadIdx.x;
    if (i >= IMG) return;
    float s = 0.f;
    #pragma unroll
    for (int k = 0; k < 16; ++k) s += tileErr[i*16 + k];
    out[i] = s * (1.0f / 256.0f);
}

// ---------------------------------------------------------------------------
extern "C" void kernel_launch(void* const* d_in, const int* in_sizes, int n_in,
                              void* d_out, int out_size, void* d_ws, size_t ws_size,
                              hipStream_t stream) {
    const float* verts     = (const float*)d_in[0];   // (1,1000,3) f32
    const int*   faces     = (const int*)  d_in[1];   // (1,1000,3) i32
    const float* q         = (const float*)d_in[2];   // (4,)
    const float* t         = (const float*)d_in[3];   // (3,)
    const float* K         = (const float*)d_in[4];   // (3,3)
    const float* image_ref = (const float*)d_in[5];   // (256,256)

    float* ws      = (float*)d_ws;
    float* v2d     = ws + WS_V2D;
    float* tileErr = ws + WS_TERR;
    float* out     = (float*)d_out;                   // 256 floats

    k_transform<<<(NV + 255)/256, 256, 0, stream>>>(verts, q, t, K, v2d);
    k_edges    <<<(NFP + 255)/256, 256, 0, stream>>>(faces, v2d, ws);
    k_render   <<<IMG * (IMG/16), 128, 0, stream>>>(ws, image_ref, tileErr);
    k_finalize <<<1, 256, 0, stream>>>(tileErr, out);
}